// ViT_Reconstruction_Model_84155589198750
// MI455X (gfx1250) — compile-verified
//
#include <hip/hip_runtime.h>
#include <hip/hip_bf16.h>
#include <math.h>

typedef float v2f __attribute__((ext_vector_type(2)));
typedef float v8f __attribute__((ext_vector_type(8)));

#define BN 64
#define TN 196
#define DE 768
#define DQ 512
#define DD 256
#define ROWS (BN * TN) /* 12544 */
#define NCODES_H 4096
#define NCODES_L 256
#define LN_EPS_C 1e-5f
#define COMMIT 0.25f
#define THRESH 0.6f

// -------------------------------------------------------------------------
// Shared WMMA inner loop: one wave accumulates a 16x64 tile of
//   acc += Atile(16 x K, staged in LDS) @ B(K x ldb, row-major, cols n0..n0+63)
// K is consumed in chunks of 16: all A/B fragments of a chunk are loaded
// into independent registers first (load clause, incremental waits), then
// 16 back-to-back V_WMMA_F32_16X16X4_F32 on 4 independent acc chains.
// -------------------------------------------------------------------------
template <int K, int LDK>
__device__ __forceinline__ void wmma_tile_16x64(
    const float* __restrict__ xs, const float* __restrict__ Bm, int ldb,
    int n0, int row16, int half, v8f acc[4]) {
  const float* arow = xs + row16 * LDK + 2 * half;
  const float* bbase = Bm + (long)(2 * half) * ldb + n0 + row16;
#pragma unroll 1
  for (int k = 0; k < K; k += 16) {
    v2f a[4];
#pragma unroll
    for (int t = 0; t < 4; ++t) {
      a[t][0] = arow[k + 4 * t];
      a[t][1] = arow[k + 4 * t + 1];
    }
    v2f b[4][4];
#pragma unroll
    for (int t = 0; t < 4; ++t) {
#pragma unroll
      for (int j = 0; j < 4; ++j) {
        const float* bp = bbase + (long)(k + 4 * t) * ldb + j * 16;
        b[t][j][0] = bp[0];
        b[t][j][1] = bp[ldb];
      }
    }
#pragma unroll
    for (int t = 0; t < 4; ++t) {
#pragma unroll
      for (int j = 0; j < 4; ++j) {
        acc[j] = __builtin_amdgcn_wmma_f32_16x16x4_f32(
            false, a[t], false, b[t][j], (short)0, acc[j], false, false);
      }
    }
  }
}

// -------------------------------------------------------------------------
// GEMM:  C[M x N] = A[M x K] (row-major) @ B[K x N] (row-major) + bias[N]
// Block = 256 threads = 8 waves; block owns a 16-row A tile (LDS); wave w
// owns columns [w*64, w*64+64).
// -------------------------------------------------------------------------
template <int K>
__global__ __launch_bounds__(256) void gemm_bias_kernel(
    const float* __restrict__ A, const float* __restrict__ Bm,
    const float* __restrict__ bias, float* __restrict__ C, int N) {
  constexpr int LDK = K + 4;  // pad to avoid LDS bank conflicts
  __shared__ float xs[16 * LDK];
  const int m0 = blockIdx.x * 16;
  const int tid = threadIdx.x;
  for (int i = tid; i < 16 * K; i += 256) {
    const int r = i / K, c = i - r * K;
    xs[r * LDK + c] = A[(m0 + r) * (long)K + c];
  }
  __syncthreads();
  const int wave = tid >> 5;
  const int lane = tid & 31;
  const int row16 = lane & 15;
  const int half = lane >> 4;
  const int n0 = wave * 64;
  if (n0 >= N) return;  // wave-uniform (N=256 uses 4 of 8 waves)
  v8f acc[4] = {};
  wmma_tile_16x64<K, LDK>(xs, Bm, N, n0, row16, half, acc);
#pragma unroll
  for (int j = 0; j < 4; ++j) {
    const int col = n0 + j * 16 + row16;
    const float bv = bias[col];
#pragma unroll
    for (int r = 0; r < 8; ++r) {
      C[(m0 + r + 8 * half) * (long)N + col] = acc[j][r] + bv;
    }
  }
}

// -------------------------------------------------------------------------
// 32x32 tiled transpose: in[rows][cols] -> out[cols][rows].
// Makes the codebook VQ B-loads lane-coalesced.
// -------------------------------------------------------------------------
__global__ __launch_bounds__(256) void transpose_kernel(
    const float* __restrict__ in, float* __restrict__ out, int rows,
    int cols) {
  __shared__ float t[32][33];
  const int bx = blockIdx.x * 32;  // row base
  const int by = blockIdx.y * 32;  // col base
  const int tx = threadIdx.x & 31;
  const int ty = threadIdx.x >> 5;  // 0..7
#pragma unroll
  for (int i = 0; i < 32; i += 8) {
    t[ty + i][tx] = in[(long)(bx + ty + i) * cols + by + tx];
  }
  __syncthreads();
#pragma unroll
  for (int i = 0; i < 32; i += 8) {
    out[(long)(by + ty + i) * rows + bx + tx] = t[tx][ty + i];
  }
}

// -------------------------------------------------------------------------
// Row layernorm (in place) over DQ=512, also emits ||x||^2 per row.
// -------------------------------------------------------------------------
__global__ __launch_bounds__(256) void layernorm_kernel(
    float* __restrict__ X, const float* __restrict__ g,
    const float* __restrict__ b, float* __restrict__ xnorm) {
  __shared__ float red[256];
  __shared__ float red2[256];
  const int row = blockIdx.x;
  const int tid = threadIdx.x;
  const float x0 = X[row * (long)DQ + tid];
  const float x1 = X[row * (long)DQ + tid + 256];
  red[tid] = x0 + x1;
  red2[tid] = x0 * x0 + x1 * x1;
  __syncthreads();
  for (int s = 128; s > 0; s >>= 1) {
    if (tid < s) {
      red[tid] += red[tid + s];
      red2[tid] += red2[tid + s];
    }
    __syncthreads();
  }
  const float mean = red[0] * (1.0f / DQ);
  const float var = red2[0] * (1.0f / DQ) - mean * mean;
  const float rstd = rsqrtf(var + LN_EPS_C);
  __syncthreads();
  const float y0 = (x0 - mean) * rstd * g[tid] + b[tid];
  const float y1 = (x1 - mean) * rstd * g[tid + 256] + b[tid + 256];
  X[row * (long)DQ + tid] = y0;
  X[row * (long)DQ + tid + 256] = y1;
  red[tid] = y0 * y0 + y1 * y1;
  __syncthreads();
  for (int s = 128; s > 0; s >>= 1) {
    if (tid < s) red[tid] += red[tid + s];
    __syncthreads();
  }
  if (tid == 0) xnorm[row] = red[0];
}

// -------------------------------------------------------------------------
// Codebook row norms ||c||^2  (one block per code row, K = 512).
// -------------------------------------------------------------------------
__global__ __launch_bounds__(256) void codenorm_kernel(
    const float* __restrict__ cb, float* __restrict__ cn) {
  __shared__ float red[256];
  const int row = blockIdx.x, tid = threadIdx.x;
  const float a = cb[row * (long)DQ + tid];
  const float b = cb[row * (long)DQ + tid + 256];
  red[tid] = a * a + b * b;
  __syncthreads();
  for (int s = 128; s > 0; s >>= 1) {
    if (tid < s) red[tid] += red[tid + s];
    __syncthreads();
  }
  if (tid == 0) cn[row] = red[0];
}

// -------------------------------------------------------------------------
// VQ argmin: block = 16 token rows; wave w sweeps code tiles of 64
// starting at w*64 (block covers 512 codes/iteration).  Dots via WMMA on
// the transposed codebook (coalesced B loads); argmin over
// d = ||c||^2 - 2 x.c;  ||x||^2 added back at the end.  First-index tie
// break matches jnp.argmin.
// -------------------------------------------------------------------------
__global__ __launch_bounds__(256) void vq_argmin_kernel(
    const float* __restrict__ xq, const float* __restrict__ cbT,
    const float* __restrict__ cn, const float* __restrict__ xnorm,
    int nCodes, float* __restrict__ dmin, int* __restrict__ idxOut) {
  constexpr int K = DQ, LDK = K + 4;
  __shared__ float xs[16 * LDK];
  __shared__ float wval[8 * 16];
  __shared__ int widx[8 * 16];
  const int m0 = blockIdx.x * 16;
  const int tid = threadIdx.x;
  for (int i = tid; i < 16 * K; i += 256) {
    const int r = i >> 9, c = i & (K - 1);
    xs[r * LDK + c] = xq[(m0 + r) * (long)K + c];
  }
  __syncthreads();
  const int wave = tid >> 5, lane = tid & 31;
  const int row16 = lane & 15, half = lane >> 4;
  float bestV[8];
  int bestI[8];
#pragma unroll
  for (int r = 0; r < 8; ++r) {
    bestV[r] = __builtin_inff();
    bestI[r] = 0x7fffffff;
  }
  for (int cbase = wave * 64; cbase < nCodes; cbase += 512) {
    v8f acc[4] = {};
    wmma_tile_16x64<K, LDK>(xs, cbT, nCodes, cbase, row16, half, acc);
    float cnv[4];
#pragma unroll
    for (int j = 0; j < 4; ++j) cnv[j] = cn[cbase + j * 16 + row16];
#pragma unroll
    for (int r = 0; r < 8; ++r) {
      float v = __builtin_inff();
      int id = 0x7fffffff;
#pragma unroll
      for (int j = 0; j < 4; ++j) {
        const float d = cnv[j] - 2.0f * acc[j][r];
        const int code = cbase + j * 16 + row16;
        if (d < v || (d == v && code < id)) {
          v = d;
          id = code;
        }
      }
      // min+argmin across the 16 lanes of this half (xor of bits 0..3
      // stays inside the 16-lane half)
#pragma unroll
      for (int off = 1; off < 16; off <<= 1) {
        const float ov = __shfl_xor(v, off, 32);
        const int oi = __shfl_xor(id, off, 32);
        if (ov < v || (ov == v && oi < id)) {
          v = ov;
          id = oi;
        }
      }
      if (v < bestV[r] || (v == bestV[r] && id < bestI[r])) {
        bestV[r] = v;
        bestI[r] = id;
      }
    }
  }
  if (row16 == 0) {
#pragma unroll
    for (int r = 0; r < 8; ++r) {
      wval[wave * 16 + half * 8 + r] = bestV[r];  // token M = r + 8*half
      widx[wave * 16 + half * 8 + r] = bestI[r];
    }
  }
  __syncthreads();
  if (tid < 16) {
    float v = wval[tid];
    int id = widx[tid];
    for (int w = 1; w < 8; ++w) {
      const float ov = wval[w * 16 + tid];
      const int oi = widx[w * 16 + tid];
      if (ov < v || (ov == v && oi < id)) {
        v = ov;
        id = oi;
      }
    }
    dmin[m0 + tid] = xnorm[m0 + tid] + v;
    idxOut[m0 + tid] = id;
  }
}

// -------------------------------------------------------------------------
// Assemble y = q + (sqrt(n_var)/h)*noise; per-row loss / mask arrays.
// -------------------------------------------------------------------------
__global__ __launch_bounds__(256) void assemble_kernel(
    const float* __restrict__ fim, const float* __restrict__ h_ri,
    const float* __restrict__ noise, const float* __restrict__ cbH,
    const float* __restrict__ cbL, const int* __restrict__ idxH,
    const int* __restrict__ idxL, const float* __restrict__ dminH,
    const float* __restrict__ dminL, float* __restrict__ y,
    float* __restrict__ lossH, float* __restrict__ lossL,
    float* __restrict__ maskA) {
  const int row = blockIdx.x, tid = threadIdx.x;
  const int batch = row / TN;
  const float s = 1.0f / (1.0f + expf(-fim[row]));
  const bool mask = s > THRESH;
  const float h0 = h_ri[batch * 2 + 0], h1 = h_ri[batch * 2 + 1];
  const float h = sqrtf(h0 * h0 + h1 * h1) * 0.70710678118654752f;
  const float cs = 0.031622776601683794f / h;  // sqrt(10^(-30/10)) / h
  const float* q =
      mask ? (cbH + (long)idxH[row] * DQ) : (cbL + (long)idxL[row] * DQ);
  y[row * (long)DQ + tid] = q[tid] + cs * noise[row * (long)DQ + tid];
  y[row * (long)DQ + tid + 256] =
      q[tid + 256] + cs * noise[row * (long)DQ + tid + 256];
  if (tid == 0) {
    const float lh = (1.0f + COMMIT) * dminH[row] * (1.0f / DQ);
    const float ll = (1.0f + COMMIT) * dminL[row] * (1.0f / DQ);
    lossH[row] = mask ? lh * s : 0.0f;
    lossL[row] = mask ? 0.0f : ll * (1.0f - s);
    maskA[row] = mask ? 1.0f : 0.0f;
  }
}

// -------------------------------------------------------------------------
// Deterministic single-block reduction -> vq_loss scalar.
// -------------------------------------------------------------------------
__global__ __launch_bounds__(256) void finalize_kernel(
    const float* __restrict__ lh, const float* __restrict__ ll,
    const float* __restrict__ mk, float* __restrict__ out_loss) {
  __shared__ float sh[256], sl[256], sm[256];
  const int tid = threadIdx.x;
  float a = 0.0f, b = 0.0f, c = 0.0f;
  for (int i = tid; i < ROWS; i += 256) {
    a += lh[i];
    b += ll[i];
    c += mk[i];
  }
  sh[tid] = a;
  sl[tid] = b;
  sm[tid] = c;
  __syncthreads();
  for (int s = 128; s > 0; s >>= 1) {
    if (tid < s) {
      sh[tid] += sh[tid + s];
      sl[tid] += sl[tid + s];
      sm[tid] += sm[tid + s];
    }
    __syncthreads();
  }
  if (tid == 0) {
    const float nh = fmaxf(sm[0], 1.0f);
    const float nl = fmaxf((float)ROWS - sm[0], 1.0f);
    out_loss[0] = 0.5f * (sh[0] / nh + sl[0] / nl);
  }
}

extern "C" void kernel_launch(void* const* d_in, const int* in_sizes, int n_in,
                              void* d_out, int out_size, void* d_ws,
                              size_t ws_size, hipStream_t stream) {
  const float* x_tokens = (const float*)d_in[0];
  const float* fim = (const float*)d_in[1];
  const float* W_proj = (const float*)d_in[2];
  const float* b_proj = (const float*)d_in[3];
  const float* ln_g = (const float*)d_in[4];
  const float* ln_b = (const float*)d_in[5];
  const float* cb_high = (const float*)d_in[6];
  const float* cb_low = (const float*)d_in[7];
  const float* W_dec = (const float*)d_in[8];
  const float* b_dec = (const float*)d_in[9];
  const float* h_ri = (const float*)d_in[10];
  const float* noise = (const float*)d_in[11];
  float* out = (float*)d_out;

  char* ws = (char*)d_ws;
  size_t off = 0;
  auto alloc = [&](size_t bytes) -> void* {
    void* p = ws + off;
    off += (bytes + 255) & ~(size_t)255;
    return p;
  };
  float* xq = (float*)alloc((size_t)ROWS * DQ * 4);  // xproj -> LN'd xq -> y
  float* cbTH = (float*)alloc((size_t)NCODES_H * DQ * 4);  // 512 x 4096
  float* cbTL = (float*)alloc((size_t)NCODES_L * DQ * 4);  // 512 x 256
  float* cnH = (float*)alloc((size_t)NCODES_H * 4);
  float* cnL = (float*)alloc((size_t)NCODES_L * 4);
  float* xnorm = (float*)alloc((size_t)ROWS * 4);
  float* dminH = (float*)alloc((size_t)ROWS * 4);
  float* dminL = (float*)alloc((size_t)ROWS * 4);
  int* idxH = (int*)alloc((size_t)ROWS * 4);
  int* idxL = (int*)alloc((size_t)ROWS * 4);
  float* lossH = (float*)alloc((size_t)ROWS * 4);
  float* lossL = (float*)alloc((size_t)ROWS * 4);
  float* maskA = (float*)alloc((size_t)ROWS * 4);

  codenorm_kernel<<<NCODES_H, 256, 0, stream>>>(cb_high, cnH);
  codenorm_kernel<<<NCODES_L, 256, 0, stream>>>(cb_low, cnL);
  transpose_kernel<<<dim3(NCODES_H / 32, DQ / 32), 256, 0, stream>>>(
      cb_high, cbTH, NCODES_H, DQ);
  transpose_kernel<<<dim3(NCODES_L / 32, DQ / 32), 256, 0, stream>>>(
      cb_low, cbTL, NCODES_L, DQ);
  gemm_bias_kernel<DE><<<ROWS / 16, 256, 0, stream>>>(x_tokens, W_proj, b_proj,
                                                      xq, DQ);
  layernorm_kernel<<<ROWS, 256, 0, stream>>>(xq, ln_g, ln_b, xnorm);
  vq_argmin_kernel<<<ROWS / 16, 256, 0, stream>>>(xq, cbTH, cnH, xnorm,
                                                  NCODES_H, dminH, idxH);
  vq_argmin_kernel<<<ROWS / 16, 256, 0, stream>>>(xq, cbTL, cnL, xnorm,
                                                  NCODES_L, dminL, idxL);
  assemble_kernel<<<ROWS, 256, 0, stream>>>(fim, h_ri, noise, cb_high, cb_low,
                                            idxH, idxL, dminH, dminL, xq,
                                            lossH, lossL, maskA);
  gemm_bias_kernel<DQ><<<ROWS / 16, 256, 0, stream>>>(xq, W_dec, b_dec, out,
                                                      DD);
  finalize_kernel<<<1, 256, 0, stream>>>(lossH, lossL, maskA,
                                         out + (size_t)ROWS * DD);
}